// TransformerModel_26577257628236
// MI455X (gfx1250) — compile-verified
//
#include <hip/hip_runtime.h>
#include <math.h>

#define NN 50000
#define NE 800000
#define NH 4
#define EPS 1e-5f
#define GW 8   // waves per GEMM block

typedef float v2f __attribute__((ext_vector_type(2)));
typedef float v8f __attribute__((ext_vector_type(8)));
typedef unsigned int v4u __attribute__((ext_vector_type(4)));
typedef int v4i __attribute__((ext_vector_type(4)));
typedef int v8i __attribute__((ext_vector_type(8)));

#if __has_builtin(__builtin_amdgcn_tensor_load_to_lds) && __has_builtin(__builtin_amdgcn_s_wait_tensorcnt)
#define USE_TDM 1
#endif

// ---------------------------------------------------------------------------
// Stage nelem f32 from global into LDS.
// TDM path: wave 0 issues one TENSOR_LOAD_TO_LDS (1-D tensor descriptor per
// CDNA5 D# layout: group0 = {count=1 | lds_addr | global_addr | type=2},
// group1 = {data_size=4B, tensor_dim0=nelem, tensor_dim1=1, tile_dim0=nelem,
// tile_dim1=1, stride=nelem}), then waits TENSORcnt==0.
// Caller must __syncthreads() afterwards.  Wave-uniform branch => only wave 0
// issues (TDM ignores EXEC, so we must skip at the branch level).
// ---------------------------------------------------------------------------
static __device__ __forceinline__ void stage_lds(const float* __restrict__ g,
                                                 float* __restrict__ s, int nelem) {
#ifdef USE_TDM
    if (threadIdx.x < 32) {
        const unsigned long long ga = (unsigned long long)(const void*)g;
        const unsigned int lds = (unsigned int)(unsigned long long)(void*)s;
        v4u g0;
        g0[0] = 1u;                                              // count=1 (valid user D#)
        g0[1] = lds;                                             // lds_addr [63:32]
        g0[2] = (unsigned int)(ga & 0xFFFFFFFFu);                // global_addr low
        g0[3] = (unsigned int)((ga >> 32) & 0x01FFFFFFu) | 0x80000000u; // ga hi | type=2
        const unsigned int ne = (unsigned int)nelem;
        v8i g1;
        g1[0] = (int)(2u << 16);                                 // data_size=2 (4 bytes)
        g1[1] = (int)((ne & 0xFFFFu) << 16);                     // tensor_dim0[15:0] @ bits63:48
        g1[2] = (int)(((ne >> 16) & 0xFFFFu) | (1u << 16));      // tensor_dim0 hi | tensor_dim1=1
        g1[3] = (int)((ne & 0xFFFFu) << 16);                     // tile_dim0 @ bits127:112
        g1[4] = 1;                                               // tile_dim1=1
        g1[5] = (int)ne;                                         // tensor_dim0_stride low
        g1[6] = 0;
        g1[7] = 0;
        v4i gz = {0, 0, 0, 0};
#if __clang_major__ >= 23
        v8i gz8 = {0, 0, 0, 0, 0, 0, 0, 0};
        __builtin_amdgcn_tensor_load_to_lds(g0, g1, gz, gz, gz8, 0);
#else
        __builtin_amdgcn_tensor_load_to_lds(g0, g1, gz, gz, 0);
#endif
        __builtin_amdgcn_s_wait_tensorcnt(0);
    }
#else
    for (int i = threadIdx.x; i < nelem; i += blockDim.x) s[i] = g[i];
#endif
}

// ---------------------------------------------------------------------------
// float atomic max via signed/unsigned integer ordering trick
// (amax is initialized to -INF = 0xFF800000)
// ---------------------------------------------------------------------------
static __device__ __forceinline__ void atomicMaxF(float* addr, float val) {
    if (val >= 0.0f) {
        atomicMax((int*)addr, __float_as_int(val));
    } else {
        atomicMin((unsigned int*)addr, __float_as_uint(val));
    }
}

// ---------------------------------------------------------------------------
// Zero / -inf initialization of per-layer accumulators
// ---------------------------------------------------------------------------
__global__ void init_buffers_kernel(float* __restrict__ agg, float* __restrict__ amax,
                                    float* __restrict__ den, int nagg, int nstat) {
    int idx = blockIdx.x * blockDim.x + threadIdx.x;
    if (idx < nagg) agg[idx] = 0.0f;
    if (idx < nstat) { amax[idx] = -INFINITY; den[idx] = 0.0f; }
}

// ---------------------------------------------------------------------------
// Fast dense GEMM via V_WMMA_F32_16X16X4_F32, no bounds checks.
// Requires nrows % 16 == 0, DOUT % 16 == 0.  W (DIN x DOUT) is staged into
// LDS with the TDM; one wave computes a 16-row x DOUT strip (DOUT/16
// accumulator tiles, A operand reused across column tiles).
// 32-bit A 16x4 layout: lanes 0-15 hold M rows, VGPR0/1 = K0/K1;
// lanes 16-31 = K2/K3.  B mirrors A; C/D per the ISA 16x16 f32 layout.
// ---------------------------------------------------------------------------
template <int DIN, int DOUT>
__global__ __launch_bounds__(GW * 32) void gemm_wmma_fast_kernel(
    const float* __restrict__ A, const float* __restrict__ W,
    const float* __restrict__ bias, float* __restrict__ out, int nrowtiles) {
    __shared__ float sW[DIN * DOUT];
    __shared__ float sB[DOUT];
    if (threadIdx.x < DOUT) sB[threadIdx.x] = bias[threadIdx.x];
    stage_lds(W, sW, DIN * DOUT);
    __syncthreads();

    const int wave = threadIdx.x >> 5;
    const int lane = threadIdx.x & 31;
    const int half = lane >> 4;
    const int l16  = lane & 15;
    const int t = blockIdx.x * GW + wave;     // row tile (wave-uniform guard)
    if (t >= nrowtiles) return;

    constexpr int CT = DOUT / 16;
    v8f acc[CT];
#pragma unroll
    for (int ct = 0; ct < CT; ++ct)
        acc[ct] = (v8f){0.f, 0.f, 0.f, 0.f, 0.f, 0.f, 0.f, 0.f};

    const float* __restrict__ Arow = A + (size_t)(t * 16 + l16) * DIN;
#pragma unroll
    for (int k0 = 0; k0 < DIN; k0 += 4) {
        const int ka = k0 + half * 2;
        const v2f a = *(const v2f*)(Arow + ka);     // 8B-aligned (ka even)
#pragma unroll
        for (int ct = 0; ct < CT; ++ct) {
            v2f b;
            b.x = sW[ka * DOUT + ct * 16 + l16];
            b.y = sW[(ka + 1) * DOUT + ct * 16 + l16];
            acc[ct] = __builtin_amdgcn_wmma_f32_16x16x4_f32(false, a, false, b,
                                                            (short)0, acc[ct], false, false);
        }
    }
#pragma unroll
    for (int ct = 0; ct < CT; ++ct) {
        const float bb = sB[ct * 16 + l16];
#pragma unroll
        for (int i = 0; i < 8; ++i) {
            const int row = t * 16 + i + half * 8;  // C layout: VGPR i -> M=i (+8 hi half)
            out[(size_t)row * DOUT + ct * 16 + l16] = acc[ct][i] + bb;
        }
    }
}

// ---------------------------------------------------------------------------
// Column-guarded WMMA GEMM for dout < 16 (layer-3 skip: 64 -> 8).
// nrows % 16 == 0.  Clamped indices + select instead of divergent loads.
// ---------------------------------------------------------------------------
__global__ void gemm_wmma_col_guard_kernel(const float* __restrict__ A, const float* __restrict__ W,
                                           const float* __restrict__ bias, float* __restrict__ out,
                                           int din, int dout) {
    const int lane = threadIdx.x & 31;
    const int half = lane >> 4;
    const int l16  = lane & 15;
    const int row  = blockIdx.x * 16 + l16;
    const int col  = l16;
    const bool cOK = col < dout;
    const int colc = cOK ? col : dout - 1;

    v8f acc = {0.f, 0.f, 0.f, 0.f, 0.f, 0.f, 0.f, 0.f};
    for (int k0 = 0; k0 < din; k0 += 4) {
        const int ka = k0 + half * 2;
        const v2f a = *(const v2f*)(A + (size_t)row * din + ka);
        const float b0 = W[(size_t)ka * dout + colc];
        const float b1 = W[(size_t)(ka + 1) * dout + colc];
        v2f b;
        b.x = cOK ? b0 : 0.f;
        b.y = cOK ? b1 : 0.f;
        acc = __builtin_amdgcn_wmma_f32_16x16x4_f32(false, a, false, b,
                                                    (short)0, acc, false, false);
    }
    if (cOK) {
        const float bb = bias[col];
#pragma unroll
        for (int i = 0; i < 8; ++i) {
            const int r = blockIdx.x * 16 + i + half * 8;
            out[(size_t)r * dout + col] = acc[i] + bb;
        }
    }
}

// ---------------------------------------------------------------------------
// Pass 1: alpha[e,h] = <q[dst,h,:], k[src,h,:] + e_vec> / sqrt(C); atomic max.
// e_vec recomputed on the fly from edge_attr with We staged in LDS (TDM).
// One thread per (edge, head).
// ---------------------------------------------------------------------------
__global__ void edge_alpha_kernel(const int* __restrict__ ei, const float* __restrict__ ea,
                                  const float* __restrict__ q, const float* __restrict__ k,
                                  const float* __restrict__ We,
                                  float* __restrict__ alpha, float* __restrict__ amax,
                                  int hc, int C, float rscale) {
    __shared__ float sWe[16 * 64];
    stage_lds(We, sWe, 16 * hc);
    __syncthreads();

    const int idx = blockIdx.x * blockDim.x + threadIdx.x;
    if (idx >= NE * NH) return;
    const int e = idx >> 2, h = idx & 3;
    const int src = ei[e], dst = ei[NE + e];
    if (e + 256 < NE) __builtin_prefetch(ea + (size_t)(e + 256) * 16, 0, 1);

    float eav[16];
    const float4* ea4 = (const float4*)(ea + (size_t)e * 16);
#pragma unroll
    for (int i = 0; i < 4; ++i) {
        float4 t = ea4[i];
        eav[4 * i] = t.x; eav[4 * i + 1] = t.y; eav[4 * i + 2] = t.z; eav[4 * i + 3] = t.w;
    }
    const float* qd   = q + (size_t)dst * hc + h * C;
    const float* ks   = k + (size_t)src * hc + h * C;
    const float* wcol = sWe + h * C;
    float al = 0.f;
    for (int c = 0; c < C; ++c) {
        float ev = 0.f;
#pragma unroll
        for (int j = 0; j < 16; ++j) ev += eav[j] * wcol[j * hc + c];
        al += qd[c] * (ks[c] + ev);
    }
    al *= rscale;
    alpha[idx] = al;
    atomicMaxF(&amax[dst * NH + h], al);
}

// ---------------------------------------------------------------------------
// Pass 2: alpha[e,h] = exp(alpha - amax[dst,h]); atomic-add denominator
// ---------------------------------------------------------------------------
__global__ void edge_expsum_kernel(const int* __restrict__ ei, float* __restrict__ alpha,
                                   const float* __restrict__ amax, float* __restrict__ den) {
    const int idx = blockIdx.x * blockDim.x + threadIdx.x;
    if (idx >= NE * NH) return;
    const int e = idx >> 2, h = idx & 3;
    const int dst = ei[NE + e];
    const float ex = __expf(alpha[idx] - amax[dst * NH + h]);
    alpha[idx] = ex;
    atomicAdd(&den[dst * NH + h], ex);
}

// ---------------------------------------------------------------------------
// Pass 3: agg[dst,h,:] += (v[src,h,:] + e_vec) * alpha / (den + 1e-16)
// ---------------------------------------------------------------------------
__global__ void edge_message_kernel(const int* __restrict__ ei, const float* __restrict__ ea,
                                    const float* __restrict__ v, const float* __restrict__ We,
                                    const float* __restrict__ alpha, const float* __restrict__ den,
                                    float* __restrict__ agg, int hc, int C) {
    __shared__ float sWe[16 * 64];
    stage_lds(We, sWe, 16 * hc);
    __syncthreads();

    const int idx = blockIdx.x * blockDim.x + threadIdx.x;
    if (idx >= NE * NH) return;
    const int e = idx >> 2, h = idx & 3;
    const int src = ei[e], dst = ei[NE + e];
    if (e + 256 < NE) __builtin_prefetch(ea + (size_t)(e + 256) * 16, 0, 1);

    float eav[16];
    const float4* ea4 = (const float4*)(ea + (size_t)e * 16);
#pragma unroll
    for (int i = 0; i < 4; ++i) {
        float4 t = ea4[i];
        eav[4 * i] = t.x; eav[4 * i + 1] = t.y; eav[4 * i + 2] = t.z; eav[4 * i + 3] = t.w;
    }
    const float w = alpha[idx] / (den[dst * NH + h] + 1e-16f);
    const float* vs   = v + (size_t)src * hc + h * C;
    const float* wcol = sWe + h * C;
    float* ag = agg + (size_t)dst * hc + h * C;
    for (int c = 0; c < C; ++c) {
        float ev = 0.f;
#pragma unroll
        for (int j = 0; j < 16; ++j) ev += eav[j] * wcol[j * hc + c];
        atomicAdd(&ag[c], (vs[c] + ev) * w);
    }
}

// ---------------------------------------------------------------------------
// Node combine for concat layers (hc = 64): beta gate + LayerNorm + ReLU.
// One wave per node; lane owns channels {lane, lane+32}; shfl_xor reductions.
// ---------------------------------------------------------------------------
__global__ void combine_ln_kernel(const float* __restrict__ agg, const float* __restrict__ skp,
                                  const float* __restrict__ betaW, const float* __restrict__ gam,
                                  const float* __restrict__ bet, float* __restrict__ hout, int nn) {
    const int node = blockIdx.x * blockDim.y + threadIdx.y;
    if (node >= nn) return;
    const int lane = threadIdx.x;           // 0..31
    const size_t base = (size_t)node * 64;
    const float o0 = agg[base + lane],      o1 = agg[base + 32 + lane];
    const float r0 = skp[base + lane],      r1 = skp[base + 32 + lane];

    float s = betaW[lane] * o0 + betaW[32 + lane] * o1
            + betaW[64 + lane] * r0 + betaW[96 + lane] * r1
            + betaW[128 + lane] * (o0 - r0) + betaW[160 + lane] * (o1 - r1);
#pragma unroll
    for (int m = 16; m >= 1; m >>= 1) s += __shfl_xor(s, m, 32);
    const float g = 1.0f / (1.0f + __expf(-s));
    const float h0 = g * r0 + (1.0f - g) * o0;
    const float h1 = g * r1 + (1.0f - g) * o1;

    float mu = h0 + h1;
#pragma unroll
    for (int m = 16; m >= 1; m >>= 1) mu += __shfl_xor(mu, m, 32);
    mu *= (1.0f / 64.0f);
    const float d0 = h0 - mu, d1 = h1 - mu;
    float var = d0 * d0 + d1 * d1;
#pragma unroll
    for (int m = 16; m >= 1; m >>= 1) var += __shfl_xor(var, m, 32);
    var *= (1.0f / 64.0f);
    const float rs = rsqrtf(var + EPS);
    const float y0 = d0 * rs * gam[lane]      + bet[lane];
    const float y1 = d1 * rs * gam[32 + lane] + bet[32 + lane];
    hout[base + lane]      = fmaxf(y0, 0.0f);
    hout[base + 32 + lane] = fmaxf(y1, 0.0f);
}

// ---------------------------------------------------------------------------
// Node combine for the final conv (concat=False): head-mean + beta gate.
// ---------------------------------------------------------------------------
__global__ void combine_mean_kernel(const float* __restrict__ agg, const float* __restrict__ skp,
                                    const float* __restrict__ betaW, float* __restrict__ hout, int nn) {
    const int node = blockIdx.x * blockDim.x + threadIdx.x;
    if (node >= nn) return;
    float o[8], r[8];
    float s = 0.f;
#pragma unroll
    for (int c = 0; c < 8; ++c) {
        const size_t b = (size_t)node * 32 + c;
        o[c] = 0.25f * (agg[b] + agg[b + 8] + agg[b + 16] + agg[b + 24]);
        r[c] = skp[(size_t)node * 8 + c];
        s += betaW[c] * o[c] + betaW[8 + c] * r[c] + betaW[16 + c] * (o[c] - r[c]);
    }
    const float g = 1.0f / (1.0f + __expf(-s));
#pragma unroll
    for (int c = 0; c < 8; ++c)
        hout[(size_t)node * 8 + c] = g * r[c] + (1.0f - g) * o[c];
}

// ---------------------------------------------------------------------------
// Final per-edge linear: out[e] = cat(h[dst], h[src], edge_attr) @ W + b
// W: 32x8 row major, staged into LDS (TDM).  One thread per edge.
// ---------------------------------------------------------------------------
__global__ void final_lin_kernel(const int* __restrict__ ei, const float* __restrict__ ea,
                                 const float* __restrict__ h3, const float* __restrict__ W,
                                 const float* __restrict__ b, float* __restrict__ out) {
    __shared__ float sW[32 * 8];
    __shared__ float sb[8];
    if (threadIdx.x < 8) sb[threadIdx.x] = b[threadIdx.x];
    stage_lds(W, sW, 32 * 8);
    __syncthreads();

    const int e = blockIdx.x * blockDim.x + threadIdx.x;
    if (e >= NE) return;
    const int src = ei[e], dst = ei[NE + e];
    if (e + 256 < NE) __builtin_prefetch(ea + (size_t)(e + 256) * 16, 0, 1);

    float in[32];
#pragma unroll
    for (int j = 0; j < 8; ++j) in[j]     = h3[(size_t)dst * 8 + j];
#pragma unroll
    for (int j = 0; j < 8; ++j) in[8 + j] = h3[(size_t)src * 8 + j];
    const float4* ea4 = (const float4*)(ea + (size_t)e * 16);
#pragma unroll
    for (int i = 0; i < 4; ++i) {
        float4 t = ea4[i];
        in[16 + 4 * i] = t.x; in[17 + 4 * i] = t.y; in[18 + 4 * i] = t.z; in[19 + 4 * i] = t.w;
    }
#pragma unroll
    for (int o = 0; o < 8; ++o) {
        float acc = sb[o];
#pragma unroll
        for (int j = 0; j < 32; ++j) acc += in[j] * sW[j * 8 + o];
        out[(size_t)e * 8 + o] = acc;
    }
}

// ---------------------------------------------------------------------------
// Host orchestration
// ---------------------------------------------------------------------------
static inline int cdiv(int a, int b) { return (a + b - 1) / b; }

extern "C" void kernel_launch(void* const* d_in, const int* in_sizes, int n_in,
                              void* d_out, int out_size, void* d_ws, size_t ws_size,
                              hipStream_t stream) {
    (void)in_sizes; (void)n_in; (void)out_size; (void)ws_size;

    const float* x  = (const float*)d_in[0];        // [NN,64]
    const int*   ei = (const int*)d_in[1];          // [2,NE]
    const float* ea = (const float*)d_in[2];        // [NE,16]

    // params leaves, jax pytree order (dict keys sorted):
    // per conv: beta.W, e.W, k.W, k.b, q.W, q.b, skip.W, skip.b, v.W, v.b
    struct ConvP { const float *bW, *eW, *kW, *kb, *qW, *qb, *sW, *sb, *vW, *vb; };
    auto getc = [&](int base) {
        ConvP c;
        c.bW = (const float*)d_in[base + 0]; c.eW = (const float*)d_in[base + 1];
        c.kW = (const float*)d_in[base + 2]; c.kb = (const float*)d_in[base + 3];
        c.qW = (const float*)d_in[base + 4]; c.qb = (const float*)d_in[base + 5];
        c.sW = (const float*)d_in[base + 6]; c.sb = (const float*)d_in[base + 7];
        c.vW = (const float*)d_in[base + 8]; c.vb = (const float*)d_in[base + 9];
        return c;
    };
    ConvP c0 = getc(3), c1 = getc(13), c2 = getc(23);
    const float* linW = (const float*)d_in[33];
    const float* linb = (const float*)d_in[34];
    const float* n0b  = (const float*)d_in[35];
    const float* n0g  = (const float*)d_in[36];
    const float* n1b  = (const float*)d_in[37];
    const float* n1g  = (const float*)d_in[38];

    // workspace layout (floats)
    float* ws = (float*)d_ws;
    const size_t NF = (size_t)NN * 64;              // 3.2M
    float* q     = ws;
    float* k     = ws + NF;
    float* v     = ws + 2 * NF;
    float* skp   = ws + 3 * NF;
    float* agg   = ws + 4 * NF;
    float* h     = ws + 5 * NF;                     // hidden features [NN,64]
    float* alpha = ws + 6 * NF;                     // [NE,NH] == 3.2M floats
    float* amax  = ws + 7 * NF;                     // [NN,NH]
    float* den   = amax + (size_t)NN * NH;
    float* h3    = den + (size_t)NN * NH;           // [NN,8]

    const int EH = NE * NH;
    const dim3 eb(256), eg(cdiv(EH, 256));
    const int NT = NN / 16;                         // 3125 row tiles (exact)
    const dim3 gb(GW * 32), gg(cdiv(NT, GW));

    auto run_layer = [&](const float* xin, const ConvP& cp, int hc, int C,
                         bool concat, const float* gamma, const float* beta_ln,
                         float* hout) {
        // init accumulators
        {
            int nagg = NN * hc, nstat = NN * NH;
            int n = nagg > nstat ? nagg : nstat;
            init_buffers_kernel<<<cdiv(n, 256), 256, 0, stream>>>(agg, amax, den, nagg, nstat);
        }
        // node projections (f32 WMMA, TDM-staged weights)
        if (hc == 64) {
            gemm_wmma_fast_kernel<64, 64><<<gg, gb, 0, stream>>>(xin, cp.qW, cp.qb, q, NT);
            gemm_wmma_fast_kernel<64, 64><<<gg, gb, 0, stream>>>(xin, cp.kW, cp.kb, k, NT);
            gemm_wmma_fast_kernel<64, 64><<<gg, gb, 0, stream>>>(xin, cp.vW, cp.vb, v, NT);
            gemm_wmma_fast_kernel<64, 64><<<gg, gb, 0, stream>>>(xin, cp.sW, cp.sb, skp, NT);
        } else {
            gemm_wmma_fast_kernel<64, 32><<<gg, gb, 0, stream>>>(xin, cp.qW, cp.qb, q, NT);
            gemm_wmma_fast_kernel<64, 32><<<gg, gb, 0, stream>>>(xin, cp.kW, cp.kb, k, NT);
            gemm_wmma_fast_kernel<64, 32><<<gg, gb, 0, stream>>>(xin, cp.vW, cp.vb, v, NT);
            gemm_wmma_col_guard_kernel<<<NT, 32, 0, stream>>>(xin, cp.sW, cp.sb, skp, 64, 8);
        }
        // edge passes
        const float rscale = 1.0f / sqrtf((float)C);
        edge_alpha_kernel<<<eg, eb, 0, stream>>>(ei, ea, q, k, cp.eW, alpha, amax, hc, C, rscale);
        edge_expsum_kernel<<<eg, eb, 0, stream>>>(ei, alpha, amax, den);
        edge_message_kernel<<<eg, eb, 0, stream>>>(ei, ea, v, cp.eW, alpha, den, agg, hc, C);
        // combine
        if (concat) {
            dim3 cb(32, 8);
            combine_ln_kernel<<<cdiv(NN, 8), cb, 0, stream>>>(agg, skp, cp.bW, gamma, beta_ln, hout, NN);
        } else {
            combine_mean_kernel<<<cdiv(NN, 256), 256, 0, stream>>>(agg, skp, cp.bW, hout, NN);
        }
    };

    run_layer(x, c0, 64, 16, true, n0g, n0b, h);
    run_layer(h, c1, 64, 16, true, n1g, n1b, h);
    run_layer(h, c2, 32, 8, false, nullptr, nullptr, h3);

    final_lin_kernel<<<cdiv(NE, 256), 256, 0, stream>>>(ei, ea, h3, linW, linb, (float*)d_out);
}